// GCNJK_7129645711840
// MI455X (gfx1250) — compile-verified
//
#include <hip/hip_runtime.h>
#include <hip/hip_bf16.h>

#define Nn   100000
#define Ee   1600000
#define DINc 128
#define Hc   64
#define DOUTc 40
#define EPSc 1e-5f

typedef __attribute__((ext_vector_type(2))) float v2f;
typedef __attribute__((ext_vector_type(8))) float v8f;

__device__ __forceinline__ void atomAddF(float* p, float v) {
  __hip_atomic_fetch_add(p, v, __ATOMIC_RELAXED, __HIP_MEMORY_SCOPE_AGENT);
}

// ---------------- degree / norm ----------------
__global__ void deg_init(float* dis) {
  int i = blockIdx.x * blockDim.x + threadIdx.x;
  if (i < Nn) dis[i] = 1.0f;                       // self loop
}
__global__ void deg_edges(const int* __restrict__ dst, float* __restrict__ dis) {
  int e = blockIdx.x * blockDim.x + threadIdx.x;
  if (e < Ee) atomAddF(&dis[dst[e]], 1.0f);
}
__global__ void deg_finish(float* dis) {
  int i = blockIdx.x * blockDim.x + threadIdx.x;
  if (i < Nn) dis[i] = rsqrtf(dis[i]);
}

// ---------------- GEMM: Y[Nn x NC] = X[Nn x K] @ W[K x NC], f32 WMMA ----------------
template <int K, int NC>
__global__ void gemm_wmma_f32(const float* __restrict__ X, const float* __restrict__ W,
                              float* __restrict__ Y, int total_waves) {
  int wave = (int)((blockIdx.x * blockDim.x + threadIdx.x) >> 5);
  if (wave >= total_waves) return;                 // wave-uniform
  int lane = threadIdx.x & 31;
  int hl  = lane >> 4;                             // half-wave
  int idx = lane & 15;
  constexpr int NT = NC / 16;
  int mt = wave / NT, nt = wave % NT;
  int row = mt * 16 + idx;                         // A row for this lane
  int col = nt * 16 + idx;                         // B/C col for this lane
  const float* xrow = X + (size_t)row * K;
  v8f c = {};
#pragma unroll
  for (int kk = 0; kk < K; kk += 4) {
    int k0 = kk + 2 * hl;
    v2f a, b;
    a.x = xrow[k0];
    a.y = xrow[k0 + 1];
    b.x = W[(size_t)k0 * NC + col];
    b.y = W[(size_t)(k0 + 1) * NC + col];
    c = __builtin_amdgcn_wmma_f32_16x16x4_f32(false, a, false, b, (short)0, c,
                                              false, false);
  }
  int m0 = mt * 16 + hl * 8;
#pragma unroll
  for (int v = 0; v < 8; ++v)
    Y[(size_t)(m0 + v) * NC + col] = c[v];
}

// ---------------- aggregation ----------------
// agg[i][c] = bias[c] + h[i][c]*dis[i]^2   (bias + self-loop message, no atomics)
__global__ void agg_init(const float* __restrict__ h, const float* __restrict__ dis,
                         const float* __restrict__ bias, float* __restrict__ agg) {
  long long t = (long long)blockIdx.x * blockDim.x + threadIdx.x;
  if (t >= (long long)Nn * Hc) return;
  int i = (int)(t >> 6);
  int c = (int)(t & 63);
  float ds = dis[i];
  agg[t] = bias[c] + h[t] * ds * ds;
}
// agg[dst][c] += h[src][c] * dis[src]*dis[dst]
__global__ void agg_edges(const float* __restrict__ h, const float* __restrict__ dis,
                          const int* __restrict__ src, const int* __restrict__ dst,
                          float* __restrict__ agg) {
  long long t = (long long)blockIdx.x * blockDim.x + threadIdx.x;
  int e = (int)(t >> 6);
  if (e >= Ee) return;
  int c = (int)(t & 63);
  int s = src[e], d = dst[e];
  float nv = dis[s] * dis[d];
  atomAddF(&agg[(size_t)d * Hc + c], h[(size_t)s * Hc + c] * nv);
}

// ---------------- batchnorm ----------------
__global__ void zero_stats(float* stats) {
  if (threadIdx.x < 2 * Hc) stats[threadIdx.x] = 0.0f;
}
__global__ void bn_stats(const float* __restrict__ x, float* __restrict__ stats) {
  __shared__ float ssum[256], ssq[256];
  int c = threadIdx.x & 63;
  int rg = threadIdx.x >> 6;                       // 0..3 row-groups per block
  float s = 0.0f, q = 0.0f;
  for (int i = blockIdx.x * 4 + rg; i < Nn; i += gridDim.x * 4) {
    float v = x[(size_t)i * Hc + c];
    s += v; q += v * v;
  }
  ssum[threadIdx.x] = s; ssq[threadIdx.x] = q;
  __syncthreads();
  if (threadIdx.x < 64) {
    s = ssum[c] + ssum[64 + c] + ssum[128 + c] + ssum[192 + c];
    q = ssq[c] + ssq[64 + c] + ssq[128 + c] + ssq[192 + c];
    atomAddF(&stats[c], s);
    atomAddF(&stats[Hc + c], q);
  }
}
__global__ void bn_apply_relu(float* __restrict__ x, const float* __restrict__ stats,
                              const float* __restrict__ g, const float* __restrict__ beta) {
  long long t = (long long)blockIdx.x * blockDim.x + threadIdx.x;
  if (t >= (long long)Nn * Hc) return;
  int c = (int)(t & 63);
  const float invN = 1.0f / (float)Nn;
  float m = stats[c] * invN;
  float var = stats[Hc + c] * invN - m * m;
  float y = (x[t] - m) * rsqrtf(var + EPSc) * g[c] + beta[c];
  x[t] = fmaxf(y, 0.0f);
}

// ---------------- final: JK max + linear ----------------
__global__ void pad_wf(const float* __restrict__ Wf, float* __restrict__ wf48) {
  int t = blockIdx.x * blockDim.x + threadIdx.x;   // 64*48
  if (t >= Hc * 48) return;
  int c = t / 48, j = t % 48;
  wf48[t] = (j < DOUTc) ? Wf[c * DOUTc + j] : 0.0f;
}
__global__ void jk_final_wmma(const float* __restrict__ x1, const float* __restrict__ x2,
                              const float* __restrict__ x3, const float* __restrict__ wf48,
                              const float* __restrict__ bf, float* __restrict__ out,
                              int total_waves) {
  int wave = (int)((blockIdx.x * blockDim.x + threadIdx.x) >> 5);
  if (wave >= total_waves) return;                 // wave-uniform
  int lane = threadIdx.x & 31;
  int hl = lane >> 4, idx = lane & 15;
  int mt = wave / 3, nt = wave % 3;                // 48 = 3 col tiles
  int row = mt * 16 + idx;
  int col = nt * 16 + idx;
  const float* r1 = x1 + (size_t)row * Hc;
  const float* r2 = x2 + (size_t)row * Hc;
  const float* r3 = x3 + (size_t)row * Hc;
  v8f c = {};
#pragma unroll
  for (int kk = 0; kk < Hc; kk += 4) {
    int k0 = kk + 2 * hl;
    v2f a, b;
    a.x = fmaxf(fmaxf(r1[k0], r2[k0]), r3[k0]);
    a.y = fmaxf(fmaxf(r1[k0 + 1], r2[k0 + 1]), r3[k0 + 1]);
    b.x = wf48[k0 * 48 + col];
    b.y = wf48[(k0 + 1) * 48 + col];
    c = __builtin_amdgcn_wmma_f32_16x16x4_f32(false, a, false, b, (short)0, c,
                                              false, false);
  }
  int m0 = mt * 16 + hl * 8;
  if (col < DOUTc) {
    float bias = bf[col];
#pragma unroll
    for (int v = 0; v < 8; ++v)
      out[(size_t)(m0 + v) * DOUTc + col] = c[v] + bias;
  }
}

extern "C" void kernel_launch(void* const* d_in, const int* in_sizes, int n_in,
                              void* d_out, int out_size, void* d_ws, size_t ws_size,
                              hipStream_t stream) {
  const float* node_feat = (const float*)d_in[0];
  const int*   eidx      = (const int*)d_in[1];
  const int*   src       = eidx;
  const int*   dst       = eidx + Ee;
  const float* W1 = (const float*)d_in[2];  const float* b1 = (const float*)d_in[3];
  const float* g1 = (const float*)d_in[4];  const float* be1 = (const float*)d_in[5];
  const float* W2 = (const float*)d_in[6];  const float* b2 = (const float*)d_in[7];
  const float* g2 = (const float*)d_in[8];  const float* be2 = (const float*)d_in[9];
  const float* W3 = (const float*)d_in[10]; const float* b3 = (const float*)d_in[11];
  const float* Wf = (const float*)d_in[12]; const float* bf = (const float*)d_in[13];
  float* out = (float*)d_out;

  float* ws    = (float*)d_ws;
  float* dis   = ws;                         // Nn
  float* h     = dis + Nn;                   // Nn*Hc
  float* x1    = h   + (size_t)Nn * Hc;      // Nn*Hc
  float* x2    = x1  + (size_t)Nn * Hc;      // Nn*Hc
  float* x3    = x2  + (size_t)Nn * Hc;      // Nn*Hc
  float* stats = x3  + (size_t)Nn * Hc;      // 2*Hc
  float* wf48  = stats + 2 * Hc;             // Hc*48

  const int TB = 256;
  const int nBlkN   = (Nn + TB - 1) / TB;
  const int nBlkE   = (Ee + TB - 1) / TB;
  const int nBlkNH  = (int)(((long long)Nn * Hc + TB - 1) / TB);   // 25000
  const int nBlkEH  = (int)(((long long)Ee * Hc + TB - 1) / TB);   // 400000
  const int wavesG  = (Nn / 16) * (Hc / 16);                       // 25000
  const int nBlkG   = (wavesG * 32 + TB - 1) / TB;                 // 3125
  const int wavesF  = (Nn / 16) * 3;                               // 18750
  const int nBlkF   = (wavesF * 32 + TB - 1) / TB;                 // 2344

  // normalization coefficients
  deg_init  <<<nBlkN, TB, 0, stream>>>(dis);
  deg_edges <<<nBlkE, TB, 0, stream>>>(dst, dis);
  deg_finish<<<nBlkN, TB, 0, stream>>>(dis);
  pad_wf    <<<(Hc * 48 + TB - 1) / TB, TB, 0, stream>>>(Wf, wf48);

  // ---- layer 1 ----
  gemm_wmma_f32<DINc, Hc><<<nBlkG, TB, 0, stream>>>(node_feat, W1, h, wavesG);
  agg_init  <<<nBlkNH, TB, 0, stream>>>(h, dis, b1, x1);
  agg_edges <<<nBlkEH, TB, 0, stream>>>(h, dis, src, dst, x1);
  zero_stats<<<1, 128, 0, stream>>>(stats);
  bn_stats  <<<256, TB, 0, stream>>>(x1, stats);
  bn_apply_relu<<<nBlkNH, TB, 0, stream>>>(x1, stats, g1, be1);

  // ---- layer 2 ----
  gemm_wmma_f32<Hc, Hc><<<nBlkG, TB, 0, stream>>>(x1, W2, h, wavesG);
  agg_init  <<<nBlkNH, TB, 0, stream>>>(h, dis, b2, x2);
  agg_edges <<<nBlkEH, TB, 0, stream>>>(h, dis, src, dst, x2);
  zero_stats<<<1, 128, 0, stream>>>(stats);
  bn_stats  <<<256, TB, 0, stream>>>(x2, stats);
  bn_apply_relu<<<nBlkNH, TB, 0, stream>>>(x2, stats, g2, be2);

  // ---- layer 3 (no BN/ReLU) ----
  gemm_wmma_f32<Hc, Hc><<<nBlkG, TB, 0, stream>>>(x2, W3, h, wavesG);
  agg_init  <<<nBlkNH, TB, 0, stream>>>(h, dis, b3, x3);
  agg_edges <<<nBlkEH, TB, 0, stream>>>(h, dis, src, dst, x3);

  // ---- JK max + final linear ----
  jk_final_wmma<<<nBlkF, TB, 0, stream>>>(x1, x2, x3, wf48, bf, out, wavesF);
}